// BlazeDetector_841813590282
// MI455X (gfx1250) — compile-verified
//
#include <hip/hip_runtime.h>
#include <math.h>

// ---------------------------------------------------------------------------
// BlazeFace decode + weighted-NMS for MI455X (gfx1250, wave32).
// One workgroup (256 threads = 8 wave32) per batch element.
// CDNA5 path: TDM tensor_load_to_lds stages the 896x16 f32 tile into LDS with
// pad_interval/pad_amount producing a stride-17 (bank-conflict-free) layout,
// overlapped with sigmoid scoring; s_wait_tensorcnt synchronizes.
// ---------------------------------------------------------------------------

#define NA        896
#define NC        16
#define STRIDE    17          // 16 coords + 1 pad slot (TDM-inserted)
#define MAXDET    100
#define NTHREADS  256
#define NWAVES    8
#define MIN_SCORE 0.75f
#define SUP_T     0.3f
#define EPSF      1e-20f

typedef __attribute__((ext_vector_type(4))) unsigned int u32x4;
typedef __attribute__((ext_vector_type(8))) int          i32x8;
typedef __attribute__((ext_vector_type(4))) int          i32x4;

#if defined(__has_builtin)
#  if __has_builtin(__builtin_amdgcn_tensor_load_to_lds) && \
      __has_builtin(__builtin_amdgcn_s_wait_tensorcnt)
#    define USE_TDM 1
#  else
#    define USE_TDM 0
#  endif
#else
#  define USE_TDM 0
#endif

__global__ __launch_bounds__(NTHREADS)
void blaze_wnms_kernel(const float* __restrict__ raw_box,    // (B, 896, 16)
                       const float* __restrict__ raw_score,  // (B, 896, 1)
                       const float* __restrict__ anchors,    // (896, 4)
                       float* __restrict__ out)               // (B, 100, 17)
{
    __shared__ __align__(16) float dets[NA * STRIDE]; // decoded coords, slot16 pad
    __shared__ float sact[NA];            // score while active, else -1
    __shared__ float wred[NWAVES];        // per-wave argmax value
    __shared__ int   wredi[NWAVES];       // per-wave argmax index
    __shared__ float red18[NWAVES * 18];  // cross-wave partial sums
    __shared__ float colsum[18];
    __shared__ float bcastV;
    __shared__ int   bcastI;

    const int tid  = threadIdx.x;
    const int wave = tid >> 5;
    const int lane = tid & 31;
    const int b    = blockIdx.x;

    const float* rb = raw_box   + (size_t)b * NA * NC;
    const float* rs = raw_score + (size_t)b * NA;
    float*       ob = out       + (size_t)b * MAXDET * STRIDE;

    // ---------------- Phase 1: stage raw tile into LDS (TDM DMA) ----------
#if USE_TDM
    if (wave == 0) {
        unsigned long long ga  = (unsigned long long)rb;
        unsigned           lds = (unsigned)(unsigned long long)(const void*)dets;
        // D# group 0: count=1 | lds_addr | global_addr[56:0] | type=2
        u32x4 g0 = { 1u,
                     lds,
                     (unsigned)(ga & 0xFFFFFFFFull),
                     (unsigned)((ga >> 32) & 0x1FFFFFFull) | (2u << 30) };
        // D# group 1: data_size=4B, pad_enable, pad_interval=16 DWORDs (enc 3),
        // pad_amount=1 DWORD (enc 0); tensor 16 x 896, tile 16 x 896, stride0=16.
        i32x8 g1 = { (int)((2u << 16) | (1u << 20) | (3u << 22)),
                     (int)(16u  << 16),    // tensor_dim0 = 16  (bits 79:48)
                     (int)(896u << 16),    // tensor_dim1 = 896 (bits 111:80)
                     (int)(16u  << 16),    // tile_dim0  = 16   (bits 127:112)
                     (int)896,             // tile_dim1  = 896  (bits 143:128)
                     (int)16,              // tensor_dim0_stride = 16 (bits 207:160)
                     0, 0 };
        i32x4 g2 = { 0, 0, 0, 0 };        // 2D tensor: groups 2/3 unused
        i32x4 g3 = { 0, 0, 0, 0 };
#if defined(__clang_major__) && (__clang_major__ >= 23)
        i32x8 g4 = { 0, 0, 0, 0, 0, 0, 0, 0 };
        __builtin_amdgcn_tensor_load_to_lds(g0, g1, g2, g3, g4, 0);
#else
        __builtin_amdgcn_tensor_load_to_lds(g0, g1, g2, g3, 0);
#endif
    }
#endif

    // Overlap with the DMA: sigmoid(clip(score)) and activity mask.
    for (int a = tid; a < NA; a += NTHREADS) {
        float x = rs[a];
        x = fminf(fmaxf(x, -100.0f), 100.0f);
        float s = 1.0f / (1.0f + __expf(-x));
        sact[a] = (s >= MIN_SCORE) ? s : -1.0f;
    }

#if USE_TDM
    if (wave == 0) __builtin_amdgcn_s_wait_tensorcnt(0);
#else
    for (int a = tid; a < NA; a += NTHREADS)
        for (int c = 0; c < NC; ++c)
            dets[a * STRIDE + c] = rb[a * NC + c];
#endif
    __syncthreads();

    // ---------------- Phase 2: decode boxes in place in LDS ---------------
    const float inv = 1.0f / 128.0f;
    for (int a = tid; a < NA; a += NTHREADS) {
        float rv[16];
#pragma unroll
        for (int c = 0; c < 16; ++c) rv[c] = dets[a * STRIDE + c];
        float ax = anchors[a * 4 + 0], ay = anchors[a * 4 + 1];
        float aw = anchors[a * 4 + 2], ah = anchors[a * 4 + 3];
        float xc = rv[0] * inv * aw + ax;
        float yc = rv[1] * inv * ah + ay;
        float ww = rv[2] * inv * aw;
        float hh = rv[3] * inv * ah;
        float* d = &dets[a * STRIDE];
        d[0] = yc - 0.5f * hh;
        d[1] = xc - 0.5f * ww;
        d[2] = yc + 0.5f * hh;
        d[3] = xc + 0.5f * ww;
#pragma unroll
        for (int k = 0; k < 6; ++k) {
            d[4 + 2 * k] = rv[4 + 2 * k] * inv * aw + ax;
            d[5 + 2 * k] = rv[5 + 2 * k] * inv * ah + ay;
        }
    }
    __syncthreads();

    // ---------------- Phase 3: 100 weighted-NMS steps ----------------------
    for (int it = 0; it < MAXDET; ++it) {
        // --- block argmax of active scores (prefer lowest index on ties) ---
        float bv = -2.0f; int bi = 0;
        for (int a = tid; a < NA; a += NTHREADS) {
            float s = sact[a];
            if (s > bv) { bv = s; bi = a; }
        }
#pragma unroll
        for (int off = 16; off > 0; off >>= 1) {
            float ov = __shfl_xor(bv, off, 32);
            int   oi = __shfl_xor(bi, off, 32);
            if (ov > bv || (ov == bv && oi < bi)) { bv = ov; bi = oi; }
        }
        if (lane == 0) { wred[wave] = bv; wredi[wave] = bi; }
        __syncthreads();
        if (tid == 0) {
            float v = wred[0]; int ix = wredi[0];
            for (int w2 = 1; w2 < NWAVES; ++w2) {
                float ov = wred[w2]; int oi = wredi[w2];
                if (ov > v || (ov == v && oi < ix)) { v = ov; ix = oi; }
            }
            bcastV = v; bcastI = ix;
        }
        __syncthreads();
        const float bestV = bcastV;
        const int   bestI = bcastI;
        const bool  anyActive = bestV > 0.0f;

        float b0 = 0.f, b1 = 0.f, b2 = 0.f, b3 = 0.f, areaA = 0.f;
        if (anyActive) {
            b0 = dets[bestI * STRIDE + 0];
            b1 = dets[bestI * STRIDE + 1];
            b2 = dets[bestI * STRIDE + 2];
            b3 = dets[bestI * STRIDE + 3];
            areaA = (b2 - b0) * (b3 - b1);
        }

        // --- IoU, suppression, weighted partial sums over owned anchors ---
        float csum[16];
#pragma unroll
        for (int c = 0; c < 16; ++c) csum[c] = 0.0f;
        float tot = 0.0f, cnt = 0.0f;
        if (anyActive) {
            for (int a = tid; a < NA; a += NTHREADS) {
                float s = sact[a];
                if (s > 0.0f) {
                    const float* d = &dets[a * STRIDE];
                    float iy = fminf(b2, d[2]) - fmaxf(b0, d[0]);
                    float ix = fminf(b3, d[3]) - fmaxf(b1, d[1]);
                    float inter = fmaxf(iy, 0.0f) * fmaxf(ix, 0.0f);
                    float areaB = (d[2] - d[0]) * (d[3] - d[1]);
                    float un    = areaA + areaB - inter;
                    float iou   = inter / fmaxf(un, EPSF);
                    if (iou > SUP_T) {
                        tot += s; cnt += 1.0f;
#pragma unroll
                        for (int c = 0; c < 16; ++c) csum[c] += s * d[c];
                        sact[a] = -1.0f;   // suppress
                    }
                }
            }
        }

        // --- reduce 18 values: wave32 shfl_xor tree, then cross-wave LDS ---
#pragma unroll
        for (int c = 0; c < 16; ++c) {
#pragma unroll
            for (int off = 16; off > 0; off >>= 1)
                csum[c] += __shfl_xor(csum[c], off, 32);
        }
#pragma unroll
        for (int off = 16; off > 0; off >>= 1) {
            tot += __shfl_xor(tot, off, 32);
            cnt += __shfl_xor(cnt, off, 32);
        }
        if (lane == 0) {
#pragma unroll
            for (int c = 0; c < 16; ++c) red18[wave * 18 + c] = csum[c];
            red18[wave * 18 + 16] = tot;
            red18[wave * 18 + 17] = cnt;
        }
        __syncthreads();
        if (tid < 18) {
            float s = 0.0f;
            for (int w2 = 0; w2 < NWAVES; ++w2) s += red18[w2 * 18 + tid];
            colsum[tid] = s;
        }
        __syncthreads();

        // --- emit row (threads 0..16 write 17 floats) ----------------------
        if (tid < 17) {
            float val;
            if (!anyActive) {
                val = 0.0f;
            } else {
                float total = colsum[16], c2 = colsum[17];
                if (c2 > 1.0f) {
                    val = (tid < 16) ? (colsum[tid] / fmaxf(total, EPSF))
                                     : (total / fmaxf(c2, 1.0f));
                } else {
                    val = (tid < 16) ? dets[bestI * STRIDE + tid] : bestV;
                }
            }
            ob[it * STRIDE + tid] = val;
        }
        // No trailing barrier needed: sact is thread-private by ownership and
        // all LDS scratch is re-written only after the next step's barriers.
    }
}

extern "C" void kernel_launch(void* const* d_in, const int* in_sizes, int n_in,
                              void* d_out, int out_size, void* d_ws, size_t ws_size,
                              hipStream_t stream) {
    const float* raw_box   = (const float*)d_in[0];
    const float* raw_score = (const float*)d_in[1];
    const float* anchors   = (const float*)d_in[2];
    float*       out       = (float*)d_out;
    const int B = in_sizes[0] / (NA * NC);   // 2048
    blaze_wnms_kernel<<<B, NTHREADS, 0, stream>>>(raw_box, raw_score, anchors, out);
}